// ContactPredictionHead_26877905338413
// MI455X (gfx1250) — compile-verified
//
#include <hip/hip_runtime.h>

#define D_MODEL 256
#define HIDDEN  128
#define B_SZ    16
#define P_SZ    87
#define K_SZ    4
#define N_SZ    32768
// (D_MODEL ** -0.5) / TAU = (1/16) / 0.5 = 0.125
#define COMB_SCALE 0.125f
#define NEG_BIG (-3.402823466e38f)

typedef float v2f __attribute__((ext_vector_type(2)));
typedef float v8f __attribute__((ext_vector_type(8)));

// ---------------------------------------------------------------------------
// Kernel 1: human contact head.  One 128-thread block per (b,p) row.
//   h = relu(x @ W1 + b1); logit = h @ W2 + b2; prob = sigmoid(logit)
// ---------------------------------------------------------------------------
__global__ void human_head_kernel(const float* __restrict__ hq,
                                  const float* __restrict__ W1,
                                  const float* __restrict__ b1,
                                  const float* __restrict__ W2,
                                  const float* __restrict__ b2,
                                  float* __restrict__ out) {
  const int row = blockIdx.x;        // 0 .. B*P-1
  const int t   = threadIdx.x;       // 0 .. 127
  __shared__ float red[HIDDEN];

  const float* x = hq + (size_t)row * D_MODEL;
  float acc = b1[t];
#pragma unroll 8
  for (int d = 0; d < D_MODEL; ++d)
    acc = fmaf(x[d], W1[d * HIDDEN + t], acc);    // W1 is (256,128) row-major
  acc = fmaxf(acc, 0.0f);                         // ReLU
  red[t] = acc * W2[t];                           // W2 is (128,1)
  __syncthreads();
  for (int s = HIDDEN / 2; s > 0; s >>= 1) {
    if (t < s) red[t] += red[t + s];
    __syncthreads();
  }
  if (t == 0) {
    const float logit = red[0] + b2[0];
    out[row]               = 1.0f / (1.0f + __expf(-logit));  // human_probs
    out[B_SZ * P_SZ + row] = logit;                           // human_logits
  }
}

// ---------------------------------------------------------------------------
// Kernel 2: q = object_queries @ Wq + bq  -> workspace (B,K,D)
// One thread per output element; Wq column reads coalesced across d.
// ---------------------------------------------------------------------------
__global__ void qproj_kernel(const float* __restrict__ oq,
                             const float* __restrict__ Wq,
                             const float* __restrict__ bq,
                             float* __restrict__ qout) {
  const int idx = blockIdx.x * blockDim.x + threadIdx.x;  // 0 .. B*K*D-1
  const int d   = idx & (D_MODEL - 1);
  const int row = idx >> 8;                               // b*K + k
  const float* x = oq + (size_t)row * D_MODEL;
  float acc = bq[d];
#pragma unroll 8
  for (int e = 0; e < D_MODEL; ++e)
    acc = fmaf(x[e], Wq[e * D_MODEL + d], acc);           // Wq (256,256) row-major
  qout[idx] = acc;
}

// ---------------------------------------------------------------------------
// Kernel 3: scaled logits via V_WMMA_F32_16X16X4_F32.
// Block = 256 threads (8 waves), grid = (N/128, B).
// Wave w handles the 16-row feats tile starting at n0.
//   A (16x4 / step):  A[m,kk] = feats[n0+m, d0+kk]
//   B (4x16 / step):  B[kk,j] = q[j, d0+kk]   (rows j>=4 zero-padded in LDS)
//   D[m,j] accumulates feats[n0+m,:] . q[j,:] over 64 steps of K=4.
// ISA lane layout (32-bit A 16x4, and B/C/D row-striped):
//   lane L: row index = L&15, K-pair select = L>>4  ->  float2 loads.
// C/D 16x16 f32 layout: vgpr r, lane L -> element (M = r + 8*(L>>4), N = L&15).
// ---------------------------------------------------------------------------
__global__ void logits_wmma_kernel(const float* __restrict__ feats,
                                   const float* __restrict__ q,
                                   float* __restrict__ logits) {
  __shared__ float qs[16 * D_MODEL];                 // 16 KB: 4 real rows + 12 zero rows
  const int b = blockIdx.y;
  const int t = threadIdx.x;

  const float* qb = q + (size_t)b * K_SZ * D_MODEL;
  for (int i = t; i < 16 * D_MODEL; i += 256)
    qs[i] = (i < K_SZ * D_MODEL) ? qb[i] : 0.0f;
  __syncthreads();

  const int lane = t & 31;
  const int wave = t >> 5;
  const int m    = lane & 15;      // A row (n within tile) == B column (k index)
  const int hi   = lane >> 4;      // K-pair selector
  const int n0   = (blockIdx.x * 8 + wave) * 16;

  const float* arow = feats + ((size_t)b * N_SZ + (size_t)(n0 + m)) * D_MODEL + 2 * hi;
  const float* brow = qs + m * D_MODEL + 2 * hi;

  v8f c = {0.0f, 0.0f, 0.0f, 0.0f, 0.0f, 0.0f, 0.0f, 0.0f};
#pragma unroll 4
  for (int s = 0; s < D_MODEL / 4; ++s) {
    v2f a  = *(const v2f*)(arow + 4 * s);    // global_load_b64, 8B aligned
    v2f bb = *(const v2f*)(brow + 4 * s);    // ds_load_b64,     8B aligned
    c = __builtin_amdgcn_wmma_f32_16x16x4_f32(false, a, false, bb,
                                              (short)0, c, false, false);
  }

  // lane's column N = m; only k = 0..3 are real query rows.
  if (m < K_SZ) {
    float* dst = logits + ((size_t)(b * K_SZ + m)) * N_SZ + (size_t)(n0 + 8 * hi);
#pragma unroll
    for (int r = 0; r < 8; ++r)
      dst[r] = c[r] * COMB_SCALE;            // n = n0 + 8*hi + r  (contiguous)
  }
}

// ---------------------------------------------------------------------------
// Kernel 4: softmax over N + weighted xyz pooling.  One block per (b,k).
// Logits (8 MB) are L2-resident; two sweeps are ~free vs. the 512 MB stream.
// ---------------------------------------------------------------------------
__global__ void softmax_pool_kernel(const float* __restrict__ logits,
                                    const float* __restrict__ xyz,
                                    float* __restrict__ out) {
  const int bk = blockIdx.x;                // 0..63  (b*K + k)
  const int b  = bk >> 2;
  const int t  = threadIdx.x;               // 0..255
  const float* lg = logits + (size_t)bk * N_SZ;
  const float* xb = xyz + (size_t)b * N_SZ * 3;

  __shared__ float rs[256], rx[256], ry[256], rz[256];

  // Pass 1: max
  float mloc = NEG_BIG;
  for (int n = t; n < N_SZ; n += 256) mloc = fmaxf(mloc, lg[n]);
  rs[t] = mloc;
  __syncthreads();
  for (int s = 128; s > 0; s >>= 1) {
    if (t < s) rs[t] = fmaxf(rs[t], rs[t + s]);
    __syncthreads();
  }
  const float mx = rs[0];
  __syncthreads();

  // Pass 2: sum of exp + weighted coords
  float sum = 0.0f, cx = 0.0f, cy = 0.0f, cz = 0.0f;
  for (int n = t; n < N_SZ; n += 256) {
    const float w = __expf(lg[n] - mx);
    sum += w;
    cx = fmaf(w, xb[3 * n + 0], cx);
    cy = fmaf(w, xb[3 * n + 1], cy);
    cz = fmaf(w, xb[3 * n + 2], cz);
  }
  rs[t] = sum; rx[t] = cx; ry[t] = cy; rz[t] = cz;
  __syncthreads();
  for (int s = 128; s > 0; s >>= 1) {
    if (t < s) {
      rs[t] += rs[t + s]; rx[t] += rx[t + s];
      ry[t] += ry[t + s]; rz[t] += rz[t + s];
    }
    __syncthreads();
  }
  if (t == 0) {
    const float inv = 1.0f / rs[0];
    float* o = out + 2 * B_SZ * P_SZ + bk * 3;
    o[0] = rx[0] * inv;
    o[1] = ry[0] * inv;
    o[2] = rz[0] * inv;
  }
}

// ---------------------------------------------------------------------------
extern "C" void kernel_launch(void* const* d_in, const int* in_sizes, int n_in,
                              void* d_out, int out_size, void* d_ws, size_t ws_size,
                              hipStream_t stream) {
  const float* hq    = (const float*)d_in[0];   // (16,87,256)
  const float* oq    = (const float*)d_in[1];   // (16,4,256)
  const float* feats = (const float*)d_in[2];   // (16,32768,256)
  const float* xyz   = (const float*)d_in[3];   // (16,32768,3)
  const float* W1    = (const float*)d_in[4];   // (256,128)
  const float* b1    = (const float*)d_in[5];   // (128,)
  const float* W2    = (const float*)d_in[6];   // (128,1)
  const float* b2    = (const float*)d_in[7];   // (1,)
  const float* Wq    = (const float*)d_in[8];   // (256,256)
  const float* bq    = (const float*)d_in[9];   // (256,)
  float* out = (float*)d_out;

  float* ws_q      = (float*)d_ws;                       // 16*4*256 f32 = 64 KB
  float* ws_logits = ws_q + B_SZ * K_SZ * D_MODEL;       // 16*4*32768 f32 = 8 MB

  human_head_kernel<<<B_SZ * P_SZ, HIDDEN, 0, stream>>>(hq, W1, b1, W2, b2, out);
  qproj_kernel<<<(B_SZ * K_SZ * D_MODEL) / 256, 256, 0, stream>>>(oq, Wq, bq, ws_q);
  dim3 grid(N_SZ / 128, B_SZ);
  logits_wmma_kernel<<<grid, 256, 0, stream>>>(feats, ws_q, ws_logits);
  softmax_pool_kernel<<<B_SZ * K_SZ, 256, 0, stream>>>(ws_logits, xyz, out);
}